// PseudoGroupContrast_63488206569894
// MI455X (gfx1250) — compile-verified
//
#include <hip/hip_runtime.h>
#include <math.h>

#define PD 192
#define NQ 150
#define QS 50
#define TEMP_INV 2.0f

// d_ws float layout
#define WS_SUMF 0       // 3*192
#define WS_SUME 576     // 3*192
#define WS_CNT  1152    // 3
#define WS_LP   1156    // 3
#define WS_CN   1160    // 3
#define WS_CALL 1163    // 1
#define WS_TOTAL_FLOATS 1164

typedef float v2f __attribute__((ext_vector_type(2)));
typedef float v8f __attribute__((ext_vector_type(8)));

// ---------------- Pass 1: class sums + counts (global reduction) ----------------
__global__ void __launch_bounds__(192) pgc_pass1(
    const float* __restrict__ feat, const float* __restrict__ ema,
    const int* __restrict__ lbl, const float* __restrict__ msk,
    float* __restrict__ ws, int rowsPerBlock, int B) {
  int d = threadIdx.x;                 // dim 0..191
  int base = blockIdx.x * rowsPerBlock;
  float f0 = 0.f, f1 = 0.f, f2 = 0.f, e0 = 0.f, e1 = 0.f, e2 = 0.f;
  float c0 = 0.f, c1 = 0.f, c2 = 0.f;
  for (int r = 0; r < rowsPerBlock; ++r) {
    int row = base + r;
    if (row >= B) break;
    int c = lbl[row];
    float m = msk[row];
    float fv = feat[(size_t)row * PD + d] * m;
    float ev = ema[(size_t)row * PD + d] * m;
    if (c == 0)      { f0 += fv; e0 += ev; c0 += m; }
    else if (c == 1) { f1 += fv; e1 += ev; c1 += m; }
    else             { f2 += fv; e2 += ev; c2 += m; }
  }
  atomicAdd(&ws[WS_SUMF + 0 * PD + d], f0);
  atomicAdd(&ws[WS_SUMF + 1 * PD + d], f1);
  atomicAdd(&ws[WS_SUMF + 2 * PD + d], f2);
  atomicAdd(&ws[WS_SUME + 0 * PD + d], e0);
  atomicAdd(&ws[WS_SUME + 1 * PD + d], e1);
  atomicAdd(&ws[WS_SUME + 2 * PD + d], e2);
  if (d == 0) {
    atomicAdd(&ws[WS_CNT + 0], c0);
    atomicAdd(&ws[WS_CNT + 1], c1);
    atomicAdd(&ws[WS_CNT + 2], c2);
  }
}

// ---------------- Pass 2: l_pos + per-class constants (1 wave) ----------------
__global__ void pgc_pass2(float* __restrict__ ws, float* __restrict__ out) {
  int lane = threadIdx.x;  // 0..31
  float lp[3];
  #pragma unroll
  for (int cls = 0; cls < 3; ++cls) {
    float cnt = fmaxf(ws[WS_CNT + cls], 1e-6f);
    float sf = 0.f, se = 0.f;
    for (int j = lane; j < PD; j += 32) {
      float cf = ws[WS_SUMF + cls * PD + j] / cnt;
      float ce = ws[WS_SUME + cls * PD + j] / cnt;
      sf += cf * cf; se += ce * ce;
    }
    #pragma unroll
    for (int m = 16; m >= 1; m >>= 1) {
      sf += __shfl_xor(sf, m, 32);
      se += __shfl_xor(se, m, 32);
    }
    float nf = fmaxf(sqrtf(sf), 1e-12f);
    float ne = fmaxf(sqrtf(se), 1e-12f);
    float dp = 0.f;
    for (int j = lane; j < PD; j += 32) {
      float cf = ws[WS_SUMF + cls * PD + j] / (cnt * nf);
      float ce = ws[WS_SUME + cls * PD + j] / (cnt * ne);
      dp += cf * ce;
    }
    #pragma unroll
    for (int m = 16; m >= 1; m >>= 1) dp += __shfl_xor(dp, m, 32);
    lp[cls] = __expf(dp * TEMP_INV);
  }
  if (lane == 0) {
    float call = 0.f;
    #pragma unroll
    for (int cls = 0; cls < 3; ++cls) {
      float denom = lp[cls] + (float)NQ;
      float cn = (float)QS * (-__logf(1.f / denom + 1e-6f))
               + (-__logf(lp[cls] / denom + 1e-6f));
      ws[WS_LP + cls] = lp[cls];
      ws[WS_CN + cls] = cn;
      call += cn;
    }
    ws[WS_CALL] = call;
    out[0] = 0.f;
  }
}

// ---------------- Pass 3: WMMA GEMM (f32 16x16x4) + loss reduction ----------------
#define LQ_STRIDE 196   // 192 + 4 pad floats -> conflict-free b64 LDS reads
#define LQ_ROWS   160   // 150 real + 10 zero rows for the padded 10th column tile

__global__ void __launch_bounds__(256) pgc_pass3(
    const float* __restrict__ feat, const int* __restrict__ lblp,
    const float* __restrict__ mskp, const float* __restrict__ queue,
    const float* __restrict__ ws, float* __restrict__ out, int B) {
  __shared__ float ldsQ[LQ_ROWS * LQ_STRIDE];  // 125440 bytes (<320KB WGP LDS)

  int tid = threadIdx.x;
  for (int i = tid; i < LQ_ROWS * LQ_STRIDE; i += 256) {
    int row = i / LQ_STRIDE;
    int col = i - row * LQ_STRIDE;
    ldsQ[i] = (row < NQ && col < PD) ? queue[row * PD + col] : 0.f;
  }
  __syncthreads();

  int wave = tid >> 5;
  int lane = tid & 31;
  int n  = lane & 15;
  int hi = lane >> 4;
  int rowBase = (blockIdx.x * 8 + wave) * 16;
  if (rowBase >= B) return;  // wave-uniform; after the only __syncthreads

  // per-row: scale = (1/T)/max(||f||,1e-12), label, mask
  float scl[16]; int lab[16]; float mk[16];
  #pragma unroll
  for (int r = 0; r < 16; ++r) {
    const float* fr = feat + (size_t)(rowBase + r) * PD;
    float s = 0.f;
    for (int j = lane; j < PD; j += 32) { float v = fr[j]; s += v * v; }
    #pragma unroll
    for (int m = 16; m >= 1; m >>= 1) s += __shfl_xor(s, m, 32);
    scl[r] = TEMP_INV / fmaxf(sqrtf(s), 1e-12f);
    lab[r] = lblp[rowBase + r];
    mk[r]  = mskp[rowBase + r];
  }

  // GEMM: 16 rows x 160 cols, K = 192, fp32 WMMA 16x16x4
  // A frag (16x4): lane m<16 -> A[m][k],A[m][k+1]; lane m+16 -> A[m][k+2],A[m][k+3]
  // B frag (4x16): lane n<16 -> B[k][n],B[k+1][n] = Q[qb+n][k..k+1]; lanes 16-31 -> k+2,k+3
  v8f acc[10] = {};
  const float* fbase = feat + (size_t)(rowBase + n) * PD + 2 * hi;
  const float* qb = &ldsQ[n * LQ_STRIDE + 2 * hi];
  #pragma unroll 1
  for (int kk = 0; kk < PD; kk += 4) {
    v2f a = *(const v2f*)(fbase + kk);
    #pragma unroll
    for (int t = 0; t < 10; ++t) {
      v2f b = *(const v2f*)(qb + t * 16 * LQ_STRIDE + kk);
      acc[t] = __builtin_amdgcn_wmma_f32_16x16x4_f32(
          false, a, false, b, (short)0, acc[t], false, false);
    }
  }

  // loss: C layout -> VGPR r, lanes 0-15: (M=r, N=lane); lanes 16-31: (M=r+8, N=lane-16)
  float lp0 = ws[WS_LP + 0], lp1 = ws[WS_LP + 1], lp2 = ws[WS_LP + 2];
  float cn0 = ws[WS_CN + 0], cn1 = ws[WS_CN + 1], cn2 = ws[WS_CN + 2];
  float callc = ws[WS_CALL];

  float wsum = 0.f;
  #pragma unroll
  for (int r = 0; r < 8; ++r) {
    float sc = hi ? scl[r + 8] : scl[r];
    int   c  = hi ? lab[r + 8] : lab[r];
    float m  = hi ? mk[r + 8]  : mk[r];
    // exp + expsum over all 150 real queue entries
    float p = 0.f;
    #pragma unroll
    for (int t = 0; t < 10; ++t) {
      int q = t * 16 + n;
      float e = (q < NQ) ? __expf(acc[t][r] * sc) : 0.f;
      acc[t][r] = e;
      p += e;
    }
    #pragma unroll
    for (int mm = 1; mm <= 8; mm <<= 1) p += __shfl_xor(p, mm, 32);
    float lpc = (c == 0) ? lp0 : ((c == 1) ? lp1 : lp2);
    float cnc = (c == 0) ? cn0 : ((c == 1) ? cn1 : cn2);
    float denom = lpc + p;
    int qlo = c * QS;
    float tp = 0.f;
    #pragma unroll
    for (int t = 0; t < 10; ++t) {
      int q = t * 16 + n;
      if (q >= qlo && q < qlo + QS)
        tp += -__logf(acc[t][r] / denom + 1e-6f);
    }
    #pragma unroll
    for (int mm = 1; mm <= 8; mm <<= 1) tp += __shfl_xor(tp, mm, 32);
    float tema = -__logf(lpc / denom + 1e-6f);
    float rowv = (m > 1e-6f) ? (tp + tema + (callc - cnc)) : callc;
    wsum += rowv;
  }
  wsum += __shfl_xor(wsum, 16, 32);  // combine both 16-lane halves
  if (lane == 0) {
    float inv = 1.f / (51.f * (float)B);
    atomicAdd(out, wsum * inv);
  }
}

// ---------------- host ----------------
extern "C" void kernel_launch(void* const* d_in, const int* in_sizes, int n_in,
                              void* d_out, int out_size, void* d_ws, size_t ws_size,
                              hipStream_t stream) {
  const float* feat  = (const float*)d_in[0];
  const float* ema   = (const float*)d_in[1];
  const int*   lbl   = (const int*)d_in[2];
  const float* msk   = (const float*)d_in[3];
  const float* queue = (const float*)d_in[4];
  float* out = (float*)d_out;
  float* ws  = (float*)d_ws;
  int B = in_sizes[0] / PD;

  hipMemsetAsync(d_ws, 0, WS_TOTAL_FLOATS * sizeof(float), stream);

  int rowsPerBlock = 256;
  int g1 = (B + rowsPerBlock - 1) / rowsPerBlock;
  pgc_pass1<<<g1, 192, 0, stream>>>(feat, ema, lbl, msk, ws, rowsPerBlock, B);
  pgc_pass2<<<1, 32, 0, stream>>>(ws, out);
  int g3 = (B + 127) / 128;
  pgc_pass3<<<g3, 256, 0, stream>>>(feat, lbl, msk, queue, ws, out, B);
}